// Net_5334349382149
// MI455X (gfx1250) — compile-verified
//
#include <hip/hip_runtime.h>
#include <hip/hip_bf16.h>
#include <math.h>

// ---------------------------------------------------------------------------
// Sizes (compile-time constants matching the reference)
// ---------------------------------------------------------------------------
#define N_NODES   20000
#define N_EDGES   160000
#define N_CONFIG  1000
#define OP_FEAT   140
#define CFG_FEAT  24
#define HID       256
#define EMB       128
#define X0_K      (EMB + OP_FEAT)      // 268
#define LATE_K    (HID + CFG_FEAT)     // 280
#define E_TOT     (N_EDGES + N_NODES)  // 180000 (with self loops)

typedef __attribute__((ext_vector_type(2))) float v2f;
typedef __attribute__((ext_vector_type(8))) float v8f;

// ---------------------------------------------------------------------------
// Helpers
// ---------------------------------------------------------------------------
__device__ __forceinline__ void atomicMaxFloat(float* addr, float v) {
    // int view orders positive floats; uint view orders negative floats.
    if (v >= 0.0f) atomicMax((int*)addr, __float_as_int(v));
    else           atomicMin((unsigned int*)addr, __float_as_uint(v));
}

__device__ __forceinline__ float gelu_exact(float x) {
    return 0.5f * x * (1.0f + erff(x * 0.70710678118654752f));
}

// ---------------------------------------------------------------------------
// Generic fill
// ---------------------------------------------------------------------------
__global__ void k_fill(float* p, int n, float v) {
    int i = blockIdx.x * blockDim.x + threadIdx.x;
    int stride = gridDim.x * blockDim.x;
    for (; i < n; i += stride) p[i] = v;
}

__global__ void k_init_bias(float* out, const float* __restrict__ bias, int M, int Nc) {
    int i = blockIdx.x * blockDim.x + threadIdx.x;
    int stride = gridDim.x * blockDim.x;
    int tot = M * Nc;
    for (; i < tot; i += stride) out[i] = bias[i % Nc];
}

// ---------------------------------------------------------------------------
// x0 = [ renorm(embed[opcode]) | (node_feat - mean)/(std+1e-4) ]   [N, 268]
// one block (128 threads) per node
// ---------------------------------------------------------------------------
__global__ void k_build_x0(const float* __restrict__ node_feat,
                           const float* __restrict__ nf_mean,
                           const float* __restrict__ nf_std,
                           const float* __restrict__ embed,
                           const int*   __restrict__ opcode,
                           float* __restrict__ x0) {
    int row = blockIdx.x;
    int c   = threadIdx.x;              // 0..127
    __shared__ float sh[EMB];
    int op = opcode[row];
    float v = embed[op * EMB + c];
    sh[c] = v * v;
    __syncthreads();
    for (int st = 64; st > 0; st >>= 1) {
        if (c < st) sh[c] += sh[c + st];
        __syncthreads();
    }
    float nrm   = sqrtf(sh[0]);
    float scale = fminf(1.0f, 1.0f / (nrm + 1e-7f));
    float* xr = x0 + (size_t)row * X0_K;
    xr[c] = v * scale;
    for (int f = c; f < OP_FEAT; f += EMB) {
        xr[EMB + f] = (node_feat[(size_t)row * OP_FEAT + f] - nf_mean[f]) / (nf_std[f] + 1e-4f);
    }
}

// ---------------------------------------------------------------------------
// FP32 WMMA GEMM: C[M,Nout] = A[M,K] @ W[K,Nout] (+ bias per column)
// block = 256 threads (8 waves). Each wave owns a 64x16 output slab:
//   4 M-tiles (register blocked, B fragment reused across 4 WMMAs) x 1 N-tile.
// grid.x = ceil(M/64), grid.y = Nout/128.
// Out-of-range rows are CLAMPED (row M-1): C-row m depends only on A-row m,
// so garbage rows are computed but never stored -> no EXEC divergence, EXEC
// stays all-1s through every WMMA.
// All four A fragments are loaded into DISTINCT registers before the WMMA
// chain so the loads issue as one clause and retire progressively
// (s_wait_loadcnt 3..0) instead of serializing load->wmma->load.
// ISA layouts (ISA 7.12.2): A 16x4 v2f (lanes0-15 K=0,1; lanes16-31 K=2,3)
//                           B 4x16 v2f (lanes0-15 rows K=0,1; lanes16-31 K=2,3)
//                           C 16x16 v8f (VGPR v = row m0+v+8*half, col n0+l16)
// ---------------------------------------------------------------------------
__global__ void k_gemm_wmma(const float* __restrict__ A,
                            const float* __restrict__ W,
                            const float* __restrict__ bias,
                            float* __restrict__ C,
                            int M, int K, int Nout) {
    int wave = threadIdx.x >> 5;        // 0..7 -> N-tile within block
    int lane = threadIdx.x & 31;
    int half = lane >> 4;               // 0 | 1
    int l16  = lane & 15;
    int n0 = (blockIdx.y * 8 + wave) * 16;
    int m0 = blockIdx.x * 64;

    // clamped A-row pointers (pointer induction over k, no per-iter muls)
    int mr0 = m0 + l16;        if (mr0 > M - 1) mr0 = M - 1;
    int mr1 = m0 + 16 + l16;   if (mr1 > M - 1) mr1 = M - 1;
    int mr2 = m0 + 32 + l16;   if (mr2 > M - 1) mr2 = M - 1;
    int mr3 = m0 + 48 + l16;   if (mr3 > M - 1) mr3 = M - 1;
    const float* a0 = A + (size_t)mr0 * K + 2 * half;
    const float* a1 = A + (size_t)mr1 * K + 2 * half;
    const float* a2 = A + (size_t)mr2 * K + 2 * half;
    const float* a3 = A + (size_t)mr3 * K + 2 * half;

    const float* bp = W + (size_t)(2 * half) * Nout + n0 + l16;
    const size_t bstep = (size_t)4 * Nout;

    v8f acc0 = {}, acc1 = {}, acc2 = {}, acc3 = {};
    for (int k0 = 0; k0 < K; k0 += 4) {
        // ---- issue all fragment loads up-front (distinct destinations) ----
        v2f bf, af0, af1, af2, af3;
        af0.x = a0[k0]; af0.y = a0[k0 + 1];
        af1.x = a1[k0]; af1.y = a1[k0 + 1];
        af2.x = a2[k0]; af2.y = a2[k0 + 1];
        af3.x = a3[k0]; af3.y = a3[k0 + 1];
        bf.x  = bp[0];
        bf.y  = bp[Nout];
        bp += bstep;
        // ---- 4 WMMAs sharing one B fragment -------------------------------
        acc0 = __builtin_amdgcn_wmma_f32_16x16x4_f32(false, af0, false, bf, (short)0, acc0, false, false);
        acc1 = __builtin_amdgcn_wmma_f32_16x16x4_f32(false, af1, false, bf, (short)0, acc1, false, false);
        acc2 = __builtin_amdgcn_wmma_f32_16x16x4_f32(false, af2, false, bf, (short)0, acc2, false, false);
        acc3 = __builtin_amdgcn_wmma_f32_16x16x4_f32(false, af3, false, bf, (short)0, acc3, false, false);
    }

    float bval = bias ? bias[n0 + l16] : 0.0f;
    float* crow = C + n0 + l16;
#pragma unroll
    for (int v = 0; v < 8; ++v) {
        int m = m0 + v + 8 * half;
        if (m < M)      crow[(size_t)m * Nout]        = acc0[v] + bval;
        if (m + 16 < M) crow[(size_t)(m + 16) * Nout] = acc1[v] + bval;
        if (m + 32 < M) crow[(size_t)(m + 32) * Nout] = acc2[v] + bval;
        if (m + 48 < M) crow[(size_t)(m + 48) * Nout] = acc3[v] + bval;
    }
}

// ---------------------------------------------------------------------------
// Column stats (sum, sumsq) over M rows; Nc <= 256, blockDim.x == Nc
// stats[0..Nc) += sum ; stats[Nc..2Nc) += sumsq   (stats pre-zeroed)
// ---------------------------------------------------------------------------
__global__ void k_colstats(const float* __restrict__ X, int M, int Nc,
                           float* __restrict__ stats) {
    int c = threadIdx.x;
    float s = 0.0f, q = 0.0f;
    for (int r = blockIdx.x; r < M; r += gridDim.x) {
        float v = X[(size_t)r * Nc + c];
        s += v; q += v * v;
    }
    atomicAdd(stats + c, s);
    atomicAdd(stats + Nc + c, q);
}

// x = gelu( (x - mean) / sqrt(var + 1e-5) ), in place
__global__ void k_normgelu(float* X, int M, int Nc, const float* __restrict__ stats) {
    int i = blockIdx.x * blockDim.x + threadIdx.x;
    int stride = gridDim.x * blockDim.x;
    int tot = M * Nc;
    float invM = 1.0f / (float)M;
    for (; i < tot; i += stride) {
        int c = i % Nc;
        float mean = stats[c] * invM;
        float var  = stats[Nc + c] * invM - mean * mean;
        float x = (X[i] - mean) * rsqrtf(var + 1e-5f);
        X[i] = gelu_exact(x);
    }
}

// ---------------------------------------------------------------------------
// GAT edge phase. edges i<E from edge_index, else self loop (i-E).
// ---------------------------------------------------------------------------
__global__ void k_edge_scores(const float* __restrict__ xl,
                              const float* __restrict__ xr,
                              const float* __restrict__ att,
                              const int* __restrict__ eidx,
                              float* __restrict__ e, float* __restrict__ mbuf) {
    int w    = blockIdx.x * (blockDim.x >> 5) + (threadIdx.x >> 5);
    int lane = threadIdx.x & 31;
    if (w >= E_TOT) return;
    int s, d;
    if (w < N_EDGES) { s = eidx[w]; d = eidx[N_EDGES + w]; }
    else             { s = w - N_EDGES; d = s; }
    const float* pl = xl + (size_t)s * HID;
    const float* pr = xr + (size_t)d * HID;
    float acc = 0.0f;
    for (int c = lane; c < HID; c += 32) {
        float h = pl[c] + pr[c];
        h = h > 0.0f ? h : 0.2f * h;     // leaky_relu(0.2)
        acc += h * att[c];
    }
    for (int off = 16; off > 0; off >>= 1) acc += __shfl_down(acc, off, 32);
    if (lane == 0) {
        e[w] = acc;
        atomicMaxFloat(mbuf + d, acc);
    }
}

__global__ void k_edge_expsum(const int* __restrict__ eidx,
                              float* __restrict__ e,
                              const float* __restrict__ mbuf,
                              float* __restrict__ sbuf) {
    int i = blockIdx.x * blockDim.x + threadIdx.x;
    int stride = gridDim.x * blockDim.x;
    for (; i < E_TOT; i += stride) {
        int d = (i < N_EDGES) ? eidx[N_EDGES + i] : (i - N_EDGES);
        float ex = expf(e[i] - mbuf[d]);
        e[i] = ex;
        atomicAdd(sbuf + d, ex);
    }
}

// out[dst] += alpha * xl[src]; one block (256 threads) per edge
__global__ void k_edge_scatter(const float* __restrict__ xl,
                               const int* __restrict__ eidx,
                               const float* __restrict__ e,
                               const float* __restrict__ sbuf,
                               float* __restrict__ out) {
    int i = blockIdx.x;
    if (i >= E_TOT) return;
    int s, d;
    if (i < N_EDGES) { s = eidx[i]; d = eidx[N_EDGES + i]; }
    else             { s = i - N_EDGES; d = s; }
    float alpha = e[i] / (sbuf[d] + 1e-16f);
    int c = threadIdx.x;
    atomicAdd(out + (size_t)d * HID + c, alpha * xl[(size_t)s * HID + c]);
}

// ---------------------------------------------------------------------------
// Pool: sum + max per column
// ---------------------------------------------------------------------------
__global__ void k_poolstats(const float* __restrict__ X, int M,
                            float* __restrict__ psum, float* __restrict__ pmax) {
    int c = threadIdx.x;                // 256
    float s = 0.0f, m = -INFINITY;
    for (int r = blockIdx.x; r < M; r += gridDim.x) {
        float v = X[(size_t)r * HID + c];
        s += v; m = fmaxf(m, v);
    }
    atomicAdd(psum + c, s);
    atomicMaxFloat(pmax + c, m);
}

__global__ void k_pool_final(const float* __restrict__ psum,
                             const float* __restrict__ pmax,
                             float* __restrict__ pool) {
    int c = threadIdx.x;
    pool[c] = psum[c] / (float)N_NODES + pmax[c];
}

// xlate[1000, 280] = [ (cfg - mean)/(std+1e-4) (24) | pool (256) ]
__global__ void k_build_xlate(const float* __restrict__ cfg,
                              const float* __restrict__ cm,
                              const float* __restrict__ cs,
                              const float* __restrict__ pool,
                              float* __restrict__ xlate) {
    int i = blockIdx.x * blockDim.x + threadIdx.x;
    int stride = gridDim.x * blockDim.x;
    int tot = N_CONFIG * LATE_K;
    for (; i < tot; i += stride) {
        int r = i / LATE_K, c = i % LATE_K;
        float v;
        if (c < CFG_FEAT) v = (cfg[r * CFG_FEAT + c] - cm[c]) / (cs[c] + 1e-4f);
        else              v = pool[c - CFG_FEAT];
        xlate[i] = v;
    }
}

// out[r] = dot(h2[r,:128], predW) + predb ; one wave per row
__global__ void k_pred(const float* __restrict__ h2,
                       const float* __restrict__ predW,
                       const float* __restrict__ predB,
                       float* __restrict__ out) {
    int w    = blockIdx.x * (blockDim.x >> 5) + (threadIdx.x >> 5);
    int lane = threadIdx.x & 31;
    if (w >= N_CONFIG) return;
    float acc = 0.0f;
    for (int c = lane; c < (HID / 2); c += 32) acc += h2[(size_t)w * (HID / 2) + c] * predW[c];
    for (int off = 16; off > 0; off >>= 1) acc += __shfl_down(acc, off, 32);
    if (lane == 0) out[w] = acc + predB[0];
}

// ---------------------------------------------------------------------------
// Host orchestration
// ---------------------------------------------------------------------------
static inline void gemm(const float* A, const float* W, const float* bias, float* C,
                        int M, int K, int Nout, hipStream_t stream) {
    dim3 g((M + 63) / 64, Nout / 128);
    k_gemm_wmma<<<g, 256, 0, stream>>>(A, W, bias, C, M, K, Nout);
}

static inline void norm_gelu(float* X, int M, int Nc, float* stats, hipStream_t stream) {
    k_fill<<<4, 256, 0, stream>>>(stats, 2 * Nc, 0.0f);
    k_colstats<<<160, Nc, 0, stream>>>(X, M, Nc, stats);
    k_normgelu<<<2048, 256, 0, stream>>>(X, M, Nc, stats);
}

extern "C" void kernel_launch(void* const* d_in, const int* in_sizes, int n_in,
                              void* d_out, int out_size, void* d_ws, size_t ws_size,
                              hipStream_t stream) {
    const float* node_feat   = (const float*)d_in[0];
    const float* config_feat = (const float*)d_in[1];
    const float* nf_mean     = (const float*)d_in[2];
    const float* nf_std      = (const float*)d_in[3];
    const float* cf_mean     = (const float*)d_in[4];
    const float* cf_std      = (const float*)d_in[5];
    const float* embed       = (const float*)d_in[6];
    const float* early_W1    = (const float*)d_in[7];
    const float* early_W2    = (const float*)d_in[8];
    const float* gat_Wl      = (const float*)d_in[9];
    const float* gat_bl      = (const float*)d_in[10];
    const float* gat_Wr      = (const float*)d_in[11];
    const float* gat_br      = (const float*)d_in[12];
    const float* gat_att     = (const float*)d_in[13];
    const float* gat_bias    = (const float*)d_in[14];
    const float* late_W1     = (const float*)d_in[15];
    const float* late_W2     = (const float*)d_in[16];
    const float* pred_W      = (const float*)d_in[17];
    const float* pred_b      = (const float*)d_in[18];
    const int*   node_opcode = (const int*)d_in[19];
    const int*   edge_index  = (const int*)d_in[20];
    float* out = (float*)d_out;

    // ---- workspace carve-up (floats) -------------------------------------
    float* w = (float*)d_ws;
    float* x0    = w;                       w += (size_t)N_NODES * X0_K;     // 5.36M
    float* xA    = w;                       w += (size_t)N_NODES * HID;      // 5.12M
    float* xB    = w;                       w += (size_t)N_NODES * HID;
    float* xl    = w;                       w += (size_t)N_NODES * HID;
    float* xr    = w;                       w += (size_t)N_NODES * HID;
    float* ebuf  = w;                       w += E_TOT;
    float* mbuf  = w;                       w += N_NODES;
    float* sbuf  = w;                       w += N_NODES;
    float* stats = w;                       w += 2 * HID;
    float* psum  = w;                       w += HID;
    float* pmax  = w;                       w += HID;
    float* pool  = w;                       w += HID;
    float* xlate = w;                       w += (size_t)N_CONFIG * LATE_K;
    float* h1    = w;                       w += (size_t)N_CONFIG * HID;
    float* h2    = w;                       w += (size_t)N_CONFIG * (HID / 2);

    // ---- 1. node features -------------------------------------------------
    k_build_x0<<<N_NODES, EMB, 0, stream>>>(node_feat, nf_mean, nf_std, embed,
                                            node_opcode, x0);

    // ---- 2. early MLP -----------------------------------------------------
    gemm(x0, early_W1, nullptr, xA, N_NODES, X0_K, HID, stream);
    norm_gelu(xA, N_NODES, HID, stats, stream);
    gemm(xA, early_W2, nullptr, xB, N_NODES, HID, HID, stream);
    norm_gelu(xB, N_NODES, HID, stats, stream);

    float* xCur = xB;
    float* xNxt = xA;

    // ---- 3. four GATv2 layers --------------------------------------------
    for (int i = 0; i < 4; ++i) {
        const float* Wl = gat_Wl + (size_t)i * HID * HID;
        const float* Wr = gat_Wr + (size_t)i * HID * HID;
        const float* bl = gat_bl + (size_t)i * HID;
        const float* br = gat_br + (size_t)i * HID;
        const float* at = gat_att + (size_t)i * HID;
        const float* gb = gat_bias + (size_t)i * HID;

        gemm(xCur, Wl, bl, xl, N_NODES, HID, HID, stream);
        gemm(xCur, Wr, br, xr, N_NODES, HID, HID, stream);

        k_fill<<<80, 256, 0, stream>>>(mbuf, N_NODES, -INFINITY);
        k_fill<<<80, 256, 0, stream>>>(sbuf, N_NODES, 0.0f);

        k_edge_scores<<<(E_TOT + 7) / 8, 256, 0, stream>>>(xl, xr, at, edge_index,
                                                           ebuf, mbuf);
        k_edge_expsum<<<(E_TOT + 255) / 256, 256, 0, stream>>>(edge_index, ebuf,
                                                               mbuf, sbuf);
        k_init_bias<<<2048, 256, 0, stream>>>(xNxt, gb, N_NODES, HID);
        k_edge_scatter<<<E_TOT, HID, 0, stream>>>(xl, edge_index, ebuf, sbuf, xNxt);

        norm_gelu(xNxt, N_NODES, HID, stats, stream);
        float* t = xCur; xCur = xNxt; xNxt = t;
    }

    // ---- 4. mean+max pool -------------------------------------------------
    k_fill<<<1, 256, 0, stream>>>(psum, HID, 0.0f);
    k_fill<<<1, 256, 0, stream>>>(pmax, HID, -INFINITY);
    k_poolstats<<<160, HID, 0, stream>>>(xCur, N_NODES, psum, pmax);
    k_pool_final<<<1, HID, 0, stream>>>(psum, pmax, pool);

    // ---- 5. late MLP + prediction ----------------------------------------
    k_build_xlate<<<512, 256, 0, stream>>>(config_feat, cf_mean, cf_std, pool, xlate);
    gemm(xlate, late_W1, nullptr, h1, N_CONFIG, LATE_K, HID, stream);
    norm_gelu(h1, N_CONFIG, HID, stats, stream);
    gemm(h1, late_W2, nullptr, h2, N_CONFIG, HID, HID / 2, stream);
    norm_gelu(h2, N_CONFIG, HID / 2, stats, stream);
    k_pred<<<(N_CONFIG + 7) / 8, 256, 0, stream>>>(h2, pred_W, pred_b, out);
}